// GraphEncoder_70454643523874
// MI455X (gfx1250) — compile-verified
//
#include <hip/hip_runtime.h>

#define NN 50000
#define NE 600000
#define DIM 128
#define NG 128
#define NL 3

typedef __attribute__((ext_vector_type(2))) float v2f;
typedef __attribute__((ext_vector_type(8))) float v8f;

// ---------------------------------------------------------------------------
// agg = x   (init for GIN aggregation: h_in = x + sum_{j->i} x_j)
// ---------------------------------------------------------------------------
__global__ void gin_copy(float* __restrict__ dst, const float* __restrict__ src, int n4) {
    int i = blockIdx.x * blockDim.x + threadIdx.x;
    if (i < n4) ((float4*)dst)[i] = ((const float4*)src)[i];
}

// ---------------------------------------------------------------------------
// scatter-add: agg[dst[e]] += x[src[e]]  (one wave per edge, float4 per lane)
// ---------------------------------------------------------------------------
__global__ void gin_scatter(float* __restrict__ agg, const float* __restrict__ x,
                            const int* __restrict__ esrc, const int* __restrict__ edst) {
    int gid = blockIdx.x * blockDim.x + threadIdx.x;
    int e = gid >> 5;        // 32 lanes per edge: 32 * float4 = 128 floats
    int c = gid & 31;
    if (e >= NE) return;
    int s = esrc[e];
    int d = edst[e];
    float4 v = ((const float4*)(x + (size_t)s * DIM))[c];
    float* p = agg + (size_t)d * DIM + (size_t)c * 4;
    atomicAdd(p + 0, v.x);
    atomicAdd(p + 1, v.y);
    atomicAdd(p + 2, v.z);
    atomicAdd(p + 3, v.w);
}

// ---------------------------------------------------------------------------
// Fused MLP on a 64-node tile:  out = relu(hin @ W1 + b1) @ W2 + b2
// 256 threads = 8 waves; wave w owns output columns [16w, 16w+16).
// Each wave caches its 16-col slice of W (32 x v2f = 64 VGPRs) and reuses it
// across 4 row-tiles -> 4x less weight traffic, 256 f32-WMMAs per wave.
// ---------------------------------------------------------------------------
#define LDSP (DIM + 4)   // +4 floats: row stride 132 -> 16 rows hit 16 distinct banks
#define MT 4             // row tiles of 16 per block (64 rows)

__launch_bounds__(256)
__global__ void gin_mlp(const float* __restrict__ hin,
                        const float* __restrict__ W1, const float* __restrict__ b1,
                        const float* __restrict__ W2, const float* __restrict__ b2,
                        float* __restrict__ xout, int n_nodes) {
    __shared__ float s_in[16 * MT][LDSP];
    __shared__ float s_h[16 * MT][LDSP];

    const int row0 = blockIdx.x * (16 * MT);
    const int tid  = threadIdx.x;
    const int wave = tid >> 5;
    const int lane = tid & 31;
    const int hi   = lane >> 4;          // 0: lanes 0-15, 1: lanes 16-31
    const int l16  = lane & 15;
    const int n0   = wave * 16;          // this wave's output column tile

    // stage 64 x 128 input tile into LDS (float4 per thread)
    for (int i = tid; i < 16 * MT * (DIM / 4); i += 256) {
        int r = i >> 5, c = i & 31;
        int row = row0 + r;
        float4 v = make_float4(0.f, 0.f, 0.f, 0.f);
        if (row < n_nodes) v = ((const float4*)(hin + (size_t)row * DIM))[c];
        *(float4*)&s_in[r][c * 4] = v;
    }
    __syncthreads();

    // ---- GEMM 1: h = relu(in @ W1 + b1) ----
    v2f bf[32];                                   // B frags, register-resident
    #pragma unroll
    for (int j = 0; j < 32; ++j) {
        int ka = 4 * j + hi * 2;                  // B: 4x16, row striped across lanes
        bf[j][0] = W1[(size_t)ka * DIM + n0 + l16];
        bf[j][1] = W1[(size_t)(ka + 1) * DIM + n0 + l16];
    }
    const float bias1 = b1[n0 + l16];
    #pragma unroll
    for (int t = 0; t < MT; ++t) {
        v8f acc = {};
        #pragma unroll
        for (int j = 0; j < 32; ++j) {
            int ka = 4 * j + hi * 2;              // A: 16x4 f32 layout (ISA 7.12.2)
            v2f a;
            a[0] = s_in[t * 16 + l16][ka];
            a[1] = s_in[t * 16 + l16][ka + 1];
            acc = __builtin_amdgcn_wmma_f32_16x16x4_f32(false, a, false, bf[j],
                                                        (short)0, acc, false, false);
        }
        int rbase = t * 16 + hi * 8;              // D rows: VGPR r -> M = r (+8 hi lanes)
        #pragma unroll
        for (int r = 0; r < 8; ++r) {
            float v = acc[r] + bias1;
            s_h[rbase + r][n0 + l16] = v > 0.f ? v : 0.f;
        }
    }
    __syncthreads();

    // ---- GEMM 2: out = h @ W2 + b2 ----
    #pragma unroll
    for (int j = 0; j < 32; ++j) {
        int ka = 4 * j + hi * 2;
        bf[j][0] = W2[(size_t)ka * DIM + n0 + l16];
        bf[j][1] = W2[(size_t)(ka + 1) * DIM + n0 + l16];
    }
    const float bias2 = b2[n0 + l16];
    const int col = n0 + l16;
    #pragma unroll
    for (int t = 0; t < MT; ++t) {
        v8f acc = {};
        #pragma unroll
        for (int j = 0; j < 32; ++j) {
            int ka = 4 * j + hi * 2;
            v2f a;
            a[0] = s_h[t * 16 + l16][ka];
            a[1] = s_h[t * 16 + l16][ka + 1];
            acc = __builtin_amdgcn_wmma_f32_16x16x4_f32(false, a, false, bf[j],
                                                        (short)0, acc, false, false);
        }
        int rbase = t * 16 + hi * 8;
        #pragma unroll
        for (int r = 0; r < 8; ++r) {
            int grow = row0 + rbase + r;
            if (grow < n_nodes) xout[(size_t)grow * DIM + col] = acc[r] + bias2;
        }
    }
}

// ---------------------------------------------------------------------------
// global mean pool
// ---------------------------------------------------------------------------
__global__ void pool_zero(float* __restrict__ pool, float* __restrict__ cnt) {
    int i = blockIdx.x * blockDim.x + threadIdx.x;
    if (i < NG * DIM) pool[i] = 0.f;
    if (i < NG) cnt[i] = 0.f;
}

__global__ void pool_accum(const float* __restrict__ x, const int* __restrict__ batch,
                           float* __restrict__ pool, float* __restrict__ cnt) {
    int gid = blockIdx.x * blockDim.x + threadIdx.x;
    int node = gid >> 5;
    int c = gid & 31;
    if (node >= NN) return;
    int g = batch[node];
    float4 v = ((const float4*)(x + (size_t)node * DIM))[c];
    float* p = pool + (size_t)g * DIM + (size_t)c * 4;
    atomicAdd(p + 0, v.x);
    atomicAdd(p + 1, v.y);
    atomicAdd(p + 2, v.z);
    atomicAdd(p + 3, v.w);
    if (c == 0) atomicAdd(&cnt[g], 1.0f);
}

__global__ void pool_final(const float* __restrict__ pool, const float* __restrict__ cnt,
                           float* __restrict__ out) {
    int i = blockIdx.x * blockDim.x + threadIdx.x;
    if (i < NG * DIM) {
        float c = cnt[i >> 7];
        out[i] = pool[i] / fmaxf(c, 1.0f);
    }
}

// ---------------------------------------------------------------------------
extern "C" void kernel_launch(void* const* d_in, const int* in_sizes, int n_in,
                              void* d_out, int out_size, void* d_ws, size_t ws_size,
                              hipStream_t stream) {
    const float* x  = (const float*)d_in[0];
    const float* W1 = (const float*)d_in[1];
    const float* b1 = (const float*)d_in[2];
    const float* W2 = (const float*)d_in[3];
    const float* b2 = (const float*)d_in[4];
    const int* ei   = (const int*)d_in[5];
    const int* bat  = (const int*)d_in[6];
    const int* esrc = ei;
    const int* edst = ei + NE;

    float* ws   = (float*)d_ws;
    float* agg  = ws;
    float* buf0 = agg + (size_t)NN * DIM;
    float* buf1 = buf0 + (size_t)NN * DIM;
    float* pool = buf1 + (size_t)NN * DIM;
    float* cnt  = pool + NG * DIM;

    const int nd4 = NN * DIM / 4;                 // float4 count of a node buffer
    const int copyBlocks = (nd4 + 255) / 256;
    const int scatBlocks = (NE * 32 + 255) / 256;
    const int mlpBlocks  = (NN + 16 * MT - 1) / (16 * MT);

    const float* xin = x;
    float* xout = buf0;
    for (int l = 0; l < NL; ++l) {
        gin_copy<<<copyBlocks, 256, 0, stream>>>(agg, xin, nd4);
        gin_scatter<<<scatBlocks, 256, 0, stream>>>(agg, xin, esrc, edst);
        gin_mlp<<<mlpBlocks, 256, 0, stream>>>(agg,
                                               W1 + (size_t)l * DIM * DIM, b1 + l * DIM,
                                               W2 + (size_t)l * DIM * DIM, b2 + l * DIM,
                                               xout, NN);
        xin  = xout;
        xout = (xout == buf0) ? buf1 : buf0;
    }

    pool_zero<<<(NG * DIM + 255) / 256, 256, 0, stream>>>(pool, cnt);
    pool_accum<<<(NN * 32 + 255) / 256, 256, 0, stream>>>(xin, bat, pool, cnt);
    pool_final<<<(NG * DIM + 255) / 256, 256, 0, stream>>>(pool, cnt, (float*)d_out);
}